// LeakyMax_B_Plus_70832600646273
// MI455X (gfx1250) — compile-verified
//
#include <hip/hip_runtime.h>
#include <stdint.h>

// Problem dims (from reference): Y[b,j] = max_i (X[b,i] + W[j,i])
#define NB   512     // batch rows of X / Y
#define DIN  1024    // reduction dim
#define DOUT 1024    // output cols / rows of W

// Tiling
#define BM 32        // X rows per block
#define BN 64        // W rows (=output cols) per block
#define BK 32        // k-chunk (elements)
#define PITCH 36     // LDS row pitch in floats: 32 + 4 dwords of TDM padding
#define KTILES (DIN / BK)

typedef uint32_t v4u __attribute__((ext_vector_type(4)));
typedef int      v4i __attribute__((ext_vector_type(4)));
typedef int      v8i __attribute__((ext_vector_type(8)));
typedef float    v4f __attribute__((ext_vector_type(4)));

// Issue one TDM 2-D tile load: global (row-major, rowStride elements) -> LDS.
// Tile is tile_d1 rows x tile_d0 elements (f32). LDS gets pad of 4 dwords per
// 32 dwords stored => LDS row pitch = tile_d0 + 4 floats (PITCH when tile_d0=32).
__device__ __forceinline__ void tdm_load_tile_2d(uint32_t lds_off,
                                                 const float* gptr,
                                                 uint32_t tensor_d0, uint32_t tensor_d1,
                                                 uint32_t rowStride,
                                                 uint32_t tile_d0, uint32_t tile_d1) {
  uint64_t ga = (uint64_t)(uintptr_t)gptr;

  // --- D# group 0 (4 dwords): control / lds_addr / global_addr / type ---
  v4u g0;
  g0[0] = 1u;                                        // count=1 (valid), user mode
  g0[1] = lds_off;                                   // lds_addr (bytes)
  g0[2] = (uint32_t)(ga & 0xFFFFFFFFu);              // global_addr[31:0]
  g0[3] = (uint32_t)((ga >> 32) & 0x1FFFFFFu)        // global_addr[56:32]
        | (2u << 30);                                // type = 2 ("image")

  // --- D# group 1 (8 dwords): sizes / strides / pad config ---
  v8i g1;
  g1[0] = (int)((2u << 16)        // data_size = 2 -> 4 bytes
              | (1u << 20)        // pad_enable
              | (4u << 22)        // pad_interval: code 4 -> every 32 dwords
              | (3u << 25));      // pad_amount:   code 3 -> 4 dwords
  g1[1] = (int)((tensor_d0 & 0xFFFFu) << 16);                    // tensor_dim0[15:0]
  g1[2] = (int)((tensor_d0 >> 16) | ((tensor_d1 & 0xFFFFu) << 16));
  g1[3] = (int)((tensor_d1 >> 16) | (tile_d0 << 16));            // tile_dim0
  g1[4] = (int)(tile_d1);                                        // tile_dim1 (tile_dim2=0)
  g1[5] = (int)rowStride;                                        // tensor_dim0_stride[31:0]
  g1[6] = 0;                                                     // stride0[47:32], dim1_stride[15:0]
  g1[7] = 0;                                                     // dim1_stride[47:16] (unused, 2-D)

  v4i gz = {0, 0, 0, 0};
#if defined(__clang_major__) && (__clang_major__ >= 23)
  v8i gz8 = {0, 0, 0, 0, 0, 0, 0, 0};
  __builtin_amdgcn_tensor_load_to_lds(g0, g1, gz, gz, gz8, 0);
#else
  __builtin_amdgcn_tensor_load_to_lds(g0, g1, gz, gz, 0);
#endif
}

__global__ __launch_bounds__(256) void tropical_maxplus_kernel(
    const float* __restrict__ X,   // [NB][DIN]
    const float* __restrict__ W,   // [DOUT][DIN]
    float* __restrict__ Y) {       // [NB][DOUT]
  __shared__ float Xs[2][BM * PITCH];
  __shared__ float Ws[2][BN * PITCH];

  const int tid = threadIdx.x;
  const int tx  = tid & 15;        // output-col lane group (cols tx + 16*j)
  const int ty  = tid >> 4;        // output-row lane group (rows ty + 16*i)
  const int rowBase = blockIdx.y * BM;
  const int colBase = blockIdx.x * BN;

  const bool issuer = (tid < 32);  // wave 0 drives the TDM

  const float* Xg = X + (size_t)rowBase * DIN;
  const float* Wg = W + (size_t)colBase * DIN;

  // Prologue: DMA tile 0 into buffer 0.
  if (issuer) {
    tdm_load_tile_2d((uint32_t)(uintptr_t)&Xs[0][0], Xg, DIN, NB,   DIN, BK, BM);
    tdm_load_tile_2d((uint32_t)(uintptr_t)&Ws[0][0], Wg, DIN, DOUT, DIN, BK, BN);
  }

  // Per-thread LDS row bases (constant across the K loop).
  const int xr0 = (ty     ) * PITCH;
  const int xr1 = (ty + 16) * PITCH;
  const int wr0 = (tx     ) * PITCH;
  const int wr1 = (tx + 16) * PITCH;
  const int wr2 = (tx + 32) * PITCH;
  const int wr3 = (tx + 48) * PITCH;

  float acc[2][4];
#pragma unroll
  for (int i = 0; i < 2; ++i)
#pragma unroll
    for (int j = 0; j < 4; ++j) acc[i][j] = -__builtin_inff();

#pragma unroll 2
  for (int t = 0; t < KTILES; ++t) {
    const int cur = t & 1;
    if (issuer) {
      if (t + 1 < KTILES) {
        const int nxt = (t + 1) & 1;
        tdm_load_tile_2d((uint32_t)(uintptr_t)&Xs[nxt][0],
                         Xg + (size_t)(t + 1) * BK, DIN, NB,   DIN, BK, BM);
        tdm_load_tile_2d((uint32_t)(uintptr_t)&Ws[nxt][0],
                         Wg + (size_t)(t + 1) * BK, DIN, DOUT, DIN, BK, BN);
        __builtin_amdgcn_s_wait_tensorcnt(2);  // tile t's 2 DMAs done (in-order)
      } else {
        __builtin_amdgcn_s_wait_tensorcnt(0);
      }
    }
    __syncthreads();  // publish tile t's LDS to all waves

    const float* xs = &Xs[cur][0];
    const float* ws = &Ws[cur][0];
#pragma unroll
    for (int kk = 0; kk < BK; kk += 4) {
      v4f xv[2], wv[4];
      xv[0] = *(const v4f*)(xs + xr0 + kk);
      xv[1] = *(const v4f*)(xs + xr1 + kk);
      wv[0] = *(const v4f*)(ws + wr0 + kk);
      wv[1] = *(const v4f*)(ws + wr1 + kk);
      wv[2] = *(const v4f*)(ws + wr2 + kk);
      wv[3] = *(const v4f*)(ws + wr3 + kk);
#pragma unroll
      for (int i = 0; i < 2; ++i) {
#pragma unroll
        for (int j = 0; j < 4; ++j) {
          v4f s = xv[i] + wv[j];         // -> 2x v_pk_add_f32
          float a = acc[i][j];
          a = fmaxf(fmaxf(a, s.x), s.y); // -> v_max3_f32
          a = fmaxf(fmaxf(a, s.z), s.w); // -> v_max3_f32
          acc[i][j] = a;
        }
      }
    }
    __syncthreads();  // tile t fully consumed before buffer is overwritten
  }

#pragma unroll
  for (int i = 0; i < 2; ++i) {
    const int r = rowBase + ty + 16 * i;
#pragma unroll
    for (int j = 0; j < 4; ++j) {
      const int c = colBase + tx + 16 * j;
      Y[(size_t)r * DOUT + c] = acc[i][j];
    }
  }
}

extern "C" void kernel_launch(void* const* d_in, const int* in_sizes, int n_in,
                              void* d_out, int out_size, void* d_ws, size_t ws_size,
                              hipStream_t stream) {
  (void)in_sizes; (void)n_in; (void)out_size; (void)d_ws; (void)ws_size;
  const float* X = (const float*)d_in[0];   // [512][1024] f32
  const float* W = (const float*)d_in[1];   // [1024][1024] f32
  float* Y = (float*)d_out;                 // [512][1024] f32

  dim3 grid(DOUT / BN, NB / BM);            // 16 x 16 = 256 blocks
  dim3 block(256);                          // 8 waves (wave32)
  tropical_maxplus_kernel<<<grid, block, 0, stream>>>(X, W, Y);
}